// JK_83975200571652
// MI455X (gfx1250) — compile-verified
//
#include <hip/hip_runtime.h>

typedef __attribute__((ext_vector_type(2))) float v2f;
typedef __attribute__((ext_vector_type(8))) float v8f;

#define NFEAT 128
#define NHID  64
#define NCLASS 40

// ---------------- degree / normalization ----------------

__global__ void deg_kernel(const int* __restrict__ dst, float* __restrict__ deg, int E) {
    int e = blockIdx.x * blockDim.x + threadIdx.x;
    if (e < E) unsafeAtomicAdd(&deg[dst[e]], 1.0f);
}

__global__ void rsqrt_kernel(const float* __restrict__ deg, float* __restrict__ dis, int n) {
    int i = blockIdx.x * blockDim.x + threadIdx.x;
    if (i < n) dis[i] = rsqrtf(deg[i] + 1.0f);
}

// ---------------- WMMA fp32 GEMM: C[M,N] = A[M,K] @ B[K,N] (+bias) ----------------
// Block = 256 threads = 8 waves. B staged transposed (zero-padded to TILES_N*16
// cols) in LDS once per block. Each wave owns MT 16-row M-tiles and all TILES_N
// 16-col N-tiles. A fragments are double-buffered in registers (prefetch of
// step k0+4 issued before the WMMA block of step k0) so the global load-to-use
// distance spans a full k-step of MT*TILES_N v_wmma ops.

template<int TILES_N, int MT>
__global__ void wmma_gemm_kernel(const float* __restrict__ A, const float* __restrict__ B,
                                 float* __restrict__ C, const float* __restrict__ bias,
                                 int M, int N, int K) {
    __shared__ float BT[64 * 128];            // [Npad][K] transposed, max 32 KB
    const int Npad = TILES_N * 16;
    const int tid  = threadIdx.x;

    if (N != Npad) {                          // zero-pad ghost columns (head: N=40)
        for (int i = tid; i < Npad * K; i += blockDim.x) BT[i] = 0.0f;
        __syncthreads();
    }
    for (int i = tid; i < K * N; i += blockDim.x) {
        int k = i / N;
        int c = i - k * N;
        BT[c * K + k] = B[i];                 // transpose: BT[n][k] = B[k][n]
    }
    __syncthreads();

    const int wave   = blockIdx.x * (blockDim.x >> 5) + (tid >> 5);
    const int tilesM = M >> 4;
    const int tm0    = wave * MT;
    if (tm0 >= tilesM) return;

    const int lane = tid & 31;
    const int m    = lane & 15;               // A-row / B-col / C-col within tile
    const int hw   = lane >> 4;               // half-wave
    const int kb   = hw * 2;                  // k sub-offset for this half

    const float* arow[MT];
    #pragma unroll
    for (int i = 0; i < MT; ++i) {
        int tm = (tm0 + i < tilesM) ? (tm0 + i) : (tilesM - 1);   // clamp (no OOB)
        arow[i] = A + (size_t)(tm * 16 + m) * (size_t)K;
    }

    v8f acc[MT][TILES_N];
    #pragma unroll
    for (int i = 0; i < MT; ++i)
        #pragma unroll
        for (int tn = 0; tn < TILES_N; ++tn) acc[i][tn] = (v8f){};

    // prime the A double-buffer
    v2f a_cur[MT];
    #pragma unroll
    for (int i = 0; i < MT; ++i) {
        a_cur[i].x = arow[i][kb];
        a_cur[i].y = arow[i][kb + 1];
    }

    for (int k0 = 0; k0 < K; k0 += 4) {
        // prefetch next k-step's A fragments (clamped on last iteration)
        const int kn = (k0 + 4 < K) ? (k0 + 4) : k0;
        v2f a_nxt[MT];
        #pragma unroll
        for (int i = 0; i < MT; ++i) {
            a_nxt[i].x = arow[i][kn + kb];
            a_nxt[i].y = arow[i][kn + kb + 1];
        }
        // B fragments from LDS (distinct regs -> overlapped ds_loads)
        v2f b[TILES_N];
        #pragma unroll
        for (int tn = 0; tn < TILES_N; ++tn) {
            const float* bp = &BT[(tn * 16 + m) * K + k0 + kb];
            b[tn].x = bp[0];
            b[tn].y = bp[1];
        }
        #pragma unroll
        for (int i = 0; i < MT; ++i)
            #pragma unroll
            for (int tn = 0; tn < TILES_N; ++tn)
                acc[i][tn] = __builtin_amdgcn_wmma_f32_16x16x4_f32(
                                 false, a_cur[i], false, b[tn],
                                 (short)0, acc[i][tn], false, false);
        #pragma unroll
        for (int i = 0; i < MT; ++i) a_cur[i] = a_nxt[i];
    }

    #pragma unroll
    for (int i = 0; i < MT; ++i) {
        int tm = tm0 + i;
        if (tm >= tilesM) break;
        #pragma unroll
        for (int tn = 0; tn < TILES_N; ++tn) {
            int ncol = tn * 16 + m;
            if (ncol < N) {
                float bv = (bias != nullptr) ? bias[ncol] : 0.0f;
                #pragma unroll
                for (int r = 0; r < 8; ++r) {
                    int row = tm * 16 + r + hw * 8;   // C/D: vgpr r, lanes16-31 -> M=r+8
                    C[(size_t)row * (size_t)N + ncol] = acc[i][tn][r] + bv;
                }
            }
        }
    }
}

// ---------------- GCN aggregation ----------------

// a[i,f] = h[i,f] * dis[i]^2   (self-loop contribution, zero-initializes accumulator)
__global__ void selfinit_kernel(const float* __restrict__ h, const float* __restrict__ dis,
                                float* __restrict__ a, int total) {
    int t = blockIdx.x * blockDim.x + threadIdx.x;
    if (t < total) {
        int i = t >> 6;           // /NHID
        float d = dis[i];
        a[t] = h[t] * d * d;
    }
}

// per-edge scatter: a[dst,f] += h[src,f] * dis[src]*dis[dst]
__global__ void scatter_kernel(const float* __restrict__ h, const float* __restrict__ dis,
                               const int* __restrict__ src, const int* __restrict__ dst,
                               float* __restrict__ a, int total) {
    int t = blockIdx.x * blockDim.x + threadIdx.x;
    if (t < total) {
        int e = t >> 6;           // edge
        int f = t & 63;           // feature
        int s = src[e];
        int d = dst[e];
        float c = dis[s] * dis[d];
        unsafeAtomicAdd(&a[(size_t)d * NHID + f], h[(size_t)s * NHID + f] * c);
    }
}

// h1[t] = relu(a[t] + bias[f])
__global__ void bias_relu_kernel(const float* __restrict__ a, const float* __restrict__ bias,
                                 float* __restrict__ o, int total) {
    int t = blockIdx.x * blockDim.x + threadIdx.x;
    if (t < total) {
        int f = t & 63;
        o[t] = fmaxf(a[t] + bias[f], 0.0f);
    }
}

// emb[t] = max(emb[t] /*h1*/, relu(a[t] + bias[f]) /*h2*/)   (in-place JK-max)
__global__ void max_relu_kernel(const float* __restrict__ a, const float* __restrict__ bias,
                                float* __restrict__ emb, int total) {
    int t = blockIdx.x * blockDim.x + threadIdx.x;
    if (t < total) {
        int f = t & 63;
        float h2 = fmaxf(a[t] + bias[f], 0.0f);
        emb[t] = fmaxf(emb[t], h2);
    }
}

// ---------------- driver ----------------

extern "C" void kernel_launch(void* const* d_in, const int* in_sizes, int n_in,
                              void* d_out, int out_size, void* d_ws, size_t ws_size,
                              hipStream_t stream) {
    const float* x   = (const float*)d_in[0];
    const int*   ei  = (const int*)  d_in[1];
    const float* W1  = (const float*)d_in[2];
    const float* b1  = (const float*)d_in[3];
    const float* W2  = (const float*)d_in[4];
    const float* b2  = (const float*)d_in[5];
    const float* Wfc = (const float*)d_in[6];
    const float* bfc = (const float*)d_in[7];

    const int n = in_sizes[0] / NFEAT;      // 100000
    const int E = in_sizes[1] / 2;          // 1600000
    const int* src = ei;
    const int* dst = ei + E;

    float* out    = (float*)d_out;
    float* emb    = out;                          // [n, NHID]
    float* logits = out + (size_t)n * NHID;       // [n, NCLASS]

    // workspace layout
    size_t nA  = ((size_t)n + 255) & ~(size_t)255;
    float* ws  = (float*)d_ws;
    float* dis = ws;                              // [n]
    float* deg = ws + nA;                         // [n]
    float* HX  = ws + 2 * nA;                     // [n, NHID]  (pre-agg h = x@W)
    float* AG  = HX + (size_t)n * NHID;           // [n, NHID]  (aggregation accumulator)

    const int B = 256;
    const int totalNF = n * NHID;                 // 6.4M
    const int totalEF = E * NHID;                 // 102.4M
    const int MT = 2;                             // m-tiles per wave
    const int tilesM = n / 16;                    // 6250
    const int gemmBlocks = ((tilesM + MT - 1) / MT + 7) / 8;   // 8 waves/block

    // ---- normalization coefficients ----
    hipMemsetAsync(deg, 0, (size_t)n * sizeof(float), stream);
    deg_kernel<<<(E + B - 1) / B, B, 0, stream>>>(dst, deg, E);
    rsqrt_kernel<<<(n + B - 1) / B, B, 0, stream>>>(deg, dis, n);

    // ---- layer 1: h1 = relu(agg(x@W1) + b1)  (h1 lives in d_out emb region) ----
    wmma_gemm_kernel<4, MT><<<gemmBlocks, B, 0, stream>>>(x, W1, HX, nullptr, n, NHID, NFEAT);
    selfinit_kernel<<<(totalNF + B - 1) / B, B, 0, stream>>>(HX, dis, AG, totalNF);
    scatter_kernel<<<(totalEF + B - 1) / B, B, 0, stream>>>(HX, dis, src, dst, AG, totalEF);
    bias_relu_kernel<<<(totalNF + B - 1) / B, B, 0, stream>>>(AG, b1, emb, totalNF);

    // ---- layer 2: h2 = relu(agg(h1@W2) + b2); emb = max(h1, h2) in place ----
    wmma_gemm_kernel<4, MT><<<gemmBlocks, B, 0, stream>>>(emb, W2, HX, nullptr, n, NHID, NHID);
    selfinit_kernel<<<(totalNF + B - 1) / B, B, 0, stream>>>(HX, dis, AG, totalNF);
    scatter_kernel<<<(totalEF + B - 1) / B, B, 0, stream>>>(HX, dis, src, dst, AG, totalEF);
    max_relu_kernel<<<(totalNF + B - 1) / B, B, 0, stream>>>(AG, b2, emb, totalNF);

    // ---- head: logits = emb @ Wfc + bfc  (N=40 -> 3 padded tiles) ----
    wmma_gemm_kernel<3, MT><<<gemmBlocks, B, 0, stream>>>(emb, Wfc, logits, bfc, n, NCLASS, NHID);
}